// CrossingNumberLoss_32220844654992
// MI455X (gfx1250) — compile-verified
//
#include <hip/hip_runtime.h>
#include <hip/hip_bf16.h>
#include <math.h>

// ---------------------------------------------------------------------------
// CrossingNumberLoss on MI455X (gfx1250, wave32)
//
// cos = ev @ ev.T, ev: [E,2] unit vectors -> K=2 GEMM via V_WMMA_F32_16X16X4_F32
// (K padded 2->4 with zeros). Symmetry: only tiles j >= i are computed:
//   count = S_upper + (S_diagtiles - diag_hits) / 2
// Epilogue: ballot+popcount -> accumulation rides the SALU pipe (co-executes
// with WMMA/VALU); counts are wave-uniform (1 atomic per counter per wave, no shuffles).
// Row-tile index is readfirstlane'd so loop control is scalar and EXEC is
// provably all-1s around every WMMA. ev (128 KB) lives entirely in LDS.
// ---------------------------------------------------------------------------

#define N_EDGES   16384
#define N_TILES   (N_EDGES / 16)   // 1024 tiles per dimension
#define THRESHOLD 0.1f

typedef float v2f __attribute__((ext_vector_type(2)));
typedef float v8f __attribute__((ext_vector_type(8)));

// counters[0] = S_upper (strict upper-triangle tiles, full 16x16 sums)
// counters[1] = diag_hits (per-edge |dot(ev_i,ev_i)| > T)
// counters[2] = S_diagtiles (full sums of the 1024 diagonal tiles)
__global__ void zero_counters_kernel(unsigned* counters) {
    counters[0] = 0u;
    counters[1] = 0u;
    counters[2] = 0u;
}

// ---- normalized edge vectors + diagonal-hit count -------------------------
__global__ __launch_bounds__(256) void edge_vec_kernel(
    const float* __restrict__ node_pos,        // [N_NODES, 2]
    const long long* __restrict__ edge_index,  // [2, E] int64
    float* __restrict__ ev,                    // [2*E] = xs ++ ys
    unsigned* __restrict__ counters) {
    int e = blockIdx.x * blockDim.x + threadIdx.x;
    bool hit = false;
    if (e < N_EDGES) {
        long long s = edge_index[e];
        long long t = edge_index[N_EDGES + e];
        float dx = node_pos[2 * t]     - node_pos[2 * s];
        float dy = node_pos[2 * t + 1] - node_pos[2 * s + 1];
        float n  = sqrtf(dx * dx + dy * dy);
        n = fmaxf(n, 1e-6f);                 // jnp.clip(norms, 1e-6)
        float x = dx / n, y = dy / n;
        ev[e]           = x;
        ev[N_EDGES + e] = y;
        hit = fabsf(x * x + y * y) > THRESHOLD;
    }
    unsigned long long m = __ballot(hit);    // wave32: low 32 bits valid
    if ((threadIdx.x & 31) == 0)
        atomicAdd(&counters[1], (unsigned)__popcll(m));
}

// wave-uniform count of |d| > T over all 256 tile elements:
// 8x v_cmp (VALU) + 8x s_bcnt1/s_add (SALU, co-executes with WMMA)
__device__ __forceinline__ unsigned tile_count(v8f d) {
    unsigned c = 0;
#pragma unroll
    for (int r = 0; r < 8; ++r)
        c += (unsigned)__popcll(__ballot(fabsf(d[r]) > THRESHOLD));
    return c;
}

// ---- upper-triangle thresholded Gram count via WMMA -----------------------
// 128 blocks x 256 threads = 1024 waves; wave i owns row tile i and sweeps
// column tiles j = i .. N_TILES-1 with scalar (wave-uniform) loop control.
__global__ __launch_bounds__(256) void gram_count_kernel(
    const float* __restrict__ ev,          // xs ++ ys
    unsigned* __restrict__ counters) {
    extern __shared__ float smem[];        // [0,E): x  [E,2E): y

    // cooperative stage: 128-bit global loads -> 128-bit LDS stores
    {
        const float4* src = (const float4*)ev;
        float4*       dst = (float4*)smem;
        for (int i = threadIdx.x; i < (2 * N_EDGES) / 4; i += 256)
            dst[i] = src[i];
    }
    __syncthreads();

    const int lane = threadIdx.x & 31;
    // wave-uniform row tile index: forces scalar loop bounds + EXEC all-1s
    const int ti = __builtin_amdgcn_readfirstlane(
        blockIdx.x * (256 / 32) + (threadIdx.x >> 5));
    const int row0 = ti * 16;

    // A fragment (16x4 f32): lanes 0-15 VGPR0=K0=x, VGPR1=K1=y; lanes 16-31 = 0
    v2f a;
    a.x = (lane < 16) ? smem[row0 + lane]           : 0.0f;
    a.y = (lane < 16) ? smem[N_EDGES + row0 + lane] : 0.0f;

    // B fragment (4x16 f32): VGPR0 lanes 0-15 = K0 row (x of col n=lane),
    // lanes 16-31 = K1 row (y of col n=lane-16); VGPR1 (K2,K3) = 0
    const int idxB = (lane < 16) ? lane : (N_EDGES + (lane - 16));

    // diagonal tile j == ti (counted in full, corrected in finalize)
    unsigned diagtile;
    {
        v2f b; b.x = smem[idxB + ti * 16]; b.y = 0.0f;
        v8f d = {};
        d = __builtin_amdgcn_wmma_f32_16x16x4_f32(
                false, a, false, b, (short)0, d, false, false);
        diagtile = tile_count(d);
    }

    // strict upper-triangle tiles j > ti (scalar trip count)
    unsigned upper = 0;
#pragma unroll 4
    for (int j = ti + 1; j < N_TILES; ++j) {
        v2f b; b.x = smem[idxB + j * 16]; b.y = 0.0f;
        v8f d = {};
        d = __builtin_amdgcn_wmma_f32_16x16x4_f32(
                false, a, false, b, (short)0, d, false, false);
        upper += tile_count(d);
    }

    if (lane == 0) {                      // counts are wave-uniform
        atomicAdd(&counters[0], upper);
        atomicAdd(&counters[2], diagtile);
    }
}

// ---- scalar finalize ------------------------------------------------------
// sum_full = 2*S_upper + S_diagtiles  (exact symmetry of the Gram matrix)
// ref count = (sum_full - diag)/2 = S_upper + (S_diagtiles - diag)/2
__global__ void finalize_kernel(const unsigned* __restrict__ counters,
                                float* __restrict__ out) {
    float upper     = (float)counters[0];
    float diag      = (float)counters[1];
    float diagtiles = (float)counters[2];
    float count = upper + (diagtiles - diag) * 0.5f;
    float denom = (float)N_EDGES * (float)(N_EDGES - 1) * 0.5f;
    out[0] = count / denom;
}

extern "C" void kernel_launch(void* const* d_in, const int* in_sizes, int n_in,
                              void* d_out, int out_size, void* d_ws, size_t ws_size,
                              hipStream_t stream) {
    const float*     node_pos   = (const float*)d_in[0];      // [8192, 2] f32
    const long long* edge_index = (const long long*)d_in[1];  // [2, 16384] i64
    float* out = (float*)d_out;                               // [1] f32

    float*    ev       = (float*)d_ws;
    unsigned* counters = (unsigned*)((char*)d_ws + (size_t)2 * N_EDGES * sizeof(float));

    zero_counters_kernel<<<1, 1, 0, stream>>>(counters);

    edge_vec_kernel<<<N_EDGES / 256, 256, 0, stream>>>(node_pos, edge_index, ev, counters);

    const size_t lds_bytes = (size_t)2 * N_EDGES * sizeof(float);  // 128 KB < 320 KB/WGP
    gram_count_kernel<<<N_TILES / 8, 256, lds_bytes, stream>>>(ev, counters);

    finalize_kernel<<<1, 1, 0, stream>>>(counters, out);
}